// DTMJax_84138409329306
// MI455X (gfx1250) — compile-verified
//
#include <hip/hip_runtime.h>
#include <hip/hip_bf16.h>

#define T_ 4
#define D_ 1024
#define N_ 256
#define K_ 64
#define V_ 32000

typedef __attribute__((ext_vector_type(2))) float v2f;
typedef __attribute__((ext_vector_type(8))) float v8f;

__device__ __forceinline__ float jax_exp_f(float x) {
    x = fminf(fmaxf(x, -700.0f), 700.0f);
    return fmaxf(expf(x), 1e-6f);
}

// ---------------- Kernel A: eta SGLD update (one wave32 per (t,d) row) ----
__global__ void dtm_eta_kernel(const float* __restrict__ eta,
                               const float* __restrict__ alpha,
                               const float* __restrict__ CDK,
                               const float* __restrict__ noise_eta,
                               float* __restrict__ eta_out) {
    int wave = threadIdx.x >> 5;
    int lane = threadIdx.x & 31;
    int row = blockIdx.x * (blockDim.x >> 5) + wave;   // t*D + d
    if (row >= T_ * D_) return;
    int t = row / D_;
    const float* e = eta + (long)row * K_;
    float e0 = e[lane], e1 = e[lane + 32];
    float m = fmaxf(e0, e1);
    for (int off = 16; off; off >>= 1) m = fmaxf(m, __shfl_xor(m, off));
    float x0 = expf(e0 - m), x1 = expf(e1 - m);
    float s = x0 + x1;
    for (int off = 16; off; off >>= 1) s += __shfl_xor(s, off);
    float inv = 1.0f / s;
    float ne = noise_eta[row] * 0.01f;
    const float* cdk = CDK + (long)row * K_;
    const float* al  = alpha + t * K_;
    float g0 = cdk[lane]      - (float)N_ * (x0 * inv);
    float g1 = cdk[lane + 32] - (float)N_ * (x1 * inv);
    float p0 = al[lane]      - e0;   // (alpha - eta) / ETA_VAR, ETA_VAR = 1
    float p1 = al[lane + 32] - e1;
    eta_out[(long)row * K_ + lane]      = e0 + 0.005f * (g0 + p0) + ne;
    eta_out[(long)row * K_ + lane + 32] = e1 + 0.005f * (g1 + p1) + ne;
}

// ---------------- init: copy CWK, CK into output (scatter bases) ----------
__global__ void dtm_init_kernel(const float* __restrict__ CWK,
                                const float* __restrict__ CK,
                                float* __restrict__ cwk_out,
                                float* __restrict__ ck_out) {
    long i = (long)blockIdx.x * blockDim.x + threadIdx.x;
    long n4 = (long)T_ * V_ * K_ / 4;
    const float4* src = (const float4*)CWK;
    float4* dst = (float4*)cwk_out;
    long stride = (long)gridDim.x * blockDim.x;
    for (long j = i; j < n4; j += stride) dst[j] = src[j];
    if (i < T_ * K_) ck_out[i] = CK[i];
}

// ---------------- Kernel B: MH steps + scatter counters -------------------
__global__ void dtm_mh_kernel(const int* __restrict__ W,
                              const int* __restrict__ Z,
                              const int* __restrict__ pw_idx,
                              const int* __restrict__ pt_in,
                              const float* __restrict__ u_mh,
                              const float* __restrict__ phi,
                              const float* __restrict__ eta_new,
                              const float* __restrict__ CDK,
                              float* __restrict__ cdk_out,
                              float* __restrict__ cwk_out,
                              float* __restrict__ ck_out,
                              float* __restrict__ z_out) {
    __shared__ float delta[K_];
    int td = blockIdx.x;                 // t*D + d
    int t = td / D_;
    int n = threadIdx.x;                 // 0..255 == N
    if (n < K_) delta[n] = 0.0f;
    __syncthreads();

    long base = (long)td * N_;
    long i = base + n;
    int pre   = Z[i];
    int prop1 = Z[base + pw_idx[i]];
    int w     = W[i];

    const float* phit = phi + ((long)t * V_ + w) * K_;
    float a1 = jax_exp_f(phit[prop1]) / jax_exp_f(phit[pre]);
    int k1 = (u_mh[2 * i] < a1) ? prop1 : pre;

    int pt = pt_in[i];
    const float* er = eta_new + (long)td * K_;
    float a2 = jax_exp_f(er[pt]) / jax_exp_f(er[k1]);
    int z = (u_mh[2 * i + 1] < a2) ? pt : k1;

    z_out[i] = (float)z;

    // per-(t,d) delta accumulated in LDS (ds_add_f32); exact +-1 adds
    atomicAdd(&delta[z],   1.0f);
    atomicAdd(&delta[pre], -1.0f);

    // CWK: global float atomics (w spans V)
    float* cwk_row = cwk_out + ((long)t * V_ + w) * K_;
    atomicAdd(&cwk_row[z],   1.0f);
    atomicAdd(&cwk_row[pre], -1.0f);

    __syncthreads();
    if (n < K_) {
        float dv = delta[n];
        cdk_out[(long)td * K_ + n] = CDK[(long)td * K_ + n] + dv;  // private row: no atomics
        atomicAdd(&ck_out[t * K_ + n], dv);
    }
}

// ------- Kernel C: per-(t,k) column max + sum-exp over V, WMMA reduce -----
// grid = 16 blocks (t x 16-wide k tile), block = 256 (8 waves split V).
// Sum-exp: V_WMMA_F32_16X16X4_F32 with A == ones => D rows = column sums of B.
// Lane l's B values always land in column l%16, so the sum over K-rows is
// layout-agnostic. Waves merged with online-softmax rescale.
__global__ void dtm_phistat_kernel(const float* __restrict__ phi,
                                   float* __restrict__ ws) {
    __shared__ float lm[8 * 16];
    __shared__ float ls[8 * 16];
    int b = blockIdx.x;
    int t  = b >> 2;
    int k0 = (b & 3) << 4;
    int wave = threadIdx.x >> 5;
    int lane = threadIdx.x & 31;
    int col  = lane & 15;
    int half = lane >> 4;
    const int CH = V_ / 8;               // 4000 rows per wave
    int v0 = wave * CH;
    const float* p = phi + (long)t * V_ * K_ + (k0 + col);

    // pass 1: chunk max (lanes l and l+16 cover even/odd rows, then swap)
    float m = -3.4e38f;
    for (int v = v0 + half; v < v0 + CH; v += 2)
        m = fmaxf(m, p[(long)v * K_]);
    m = fmaxf(m, __shfl_xor(m, 16));

    // pass 2: sum exp via WMMA accumulate (4 rows x 16 cols per op)
    v8f acc = {0.f, 0.f, 0.f, 0.f, 0.f, 0.f, 0.f, 0.f};
    v2f a = {1.0f, 1.0f};
    for (int v = v0; v < v0 + CH; v += 4) {
        v2f bm;
        bm.x = expf(p[(long)(v + 2 * half)     * K_] - m);
        bm.y = expf(p[(long)(v + 2 * half + 1) * K_] - m);
        acc = __builtin_amdgcn_wmma_f32_16x16x4_f32(
            false, a, false, bm, (short)0, acc, false, false);
    }
    if (lane < 16) { lm[wave * 16 + col] = m; ls[wave * 16 + col] = acc[0]; }
    __syncthreads();
    if (threadIdx.x < 16) {                      // threadIdx.x == col
        float mt = lm[col];
        for (int wv = 1; wv < 8; ++wv) mt = fmaxf(mt, lm[wv * 16 + col]);
        float st = 0.0f;
        for (int wv = 0; wv < 8; ++wv) st += ls[wv * 16 + col] * expf(lm[wv * 16 + col] - mt);
        ws[t * K_ + k0 + col] = mt;
        ws[T_ * K_ + t * K_ + k0 + col] = st;
    }
}

// ---------------- Kernel D: phi SGLD update, 4-step recursion -------------
__global__ void dtm_phi_kernel(const float* __restrict__ phi,
                               const float* __restrict__ cwk_new,
                               const float* __restrict__ ck_new,
                               const float* __restrict__ noise_phi,
                               const float* __restrict__ ws,
                               float* __restrict__ phi_out) {
    const float eps = 0.01f, he = 0.005f;
    const float c0 = (float)(2.0 * (1.0 / (1.0 / 100.0 + 1.0)));  // 2*phi_sigma/PHI_VAR
    float n0 = noise_phi[0] * eps, n1 = noise_phi[1] * eps;
    float n2 = noise_phi[2] * eps, n3 = noise_phi[3] * eps;
    long total = (long)V_ * K_;
    long stride = (long)gridDim.x * blockDim.x;
    for (long j = (long)blockIdx.x * blockDim.x + threadIdx.x; j < total; j += stride) {
        int k = (int)(j & (K_ - 1));
        float p[T_], g[T_];
#pragma unroll
        for (int t = 0; t < T_; ++t) {
            long o = (long)t * total + j;
            float pv = phi[o];
            p[t] = pv;
            float M = ws[t * K_ + k];
            float S = ws[T_ * K_ + t * K_ + k];
            float sm = expf(pv - M) / S;
            g[t] = cwk_new[o] - ck_new[t * K_ + k] * sm;
        }
        float r0 = p[0] + he * (g[0] + (c0 * p[1] - 2.0f * p[0])) + n0;
        float r1 = p[1] + he * (g[1] + (p[2] + r0 - 2.0f * p[1])) + n1;
        float r2 = p[2] + he * (g[2] + (p[3] + r1 - 2.0f * p[2])) + n2;
        float r3 = p[3] + he * (g[3] + (r2 - p[3])) + n3;
        phi_out[j]             = r0;
        phi_out[total + j]     = r1;
        phi_out[2 * total + j] = r2;
        phi_out[3 * total + j] = r3;
    }
}

extern "C" void kernel_launch(void* const* d_in, const int* in_sizes, int n_in,
                              void* d_out, int out_size, void* d_ws, size_t ws_size,
                              hipStream_t stream) {
    const int*   W         = (const int*)d_in[0];
    const int*   Z         = (const int*)d_in[1];
    const int*   pw        = (const int*)d_in[2];
    const int*   pt        = (const int*)d_in[3];
    const float* u_mh      = (const float*)d_in[4];
    const float* noise_eta = (const float*)d_in[5];
    const float* noise_phi = (const float*)d_in[6];
    const float* alpha     = (const float*)d_in[7];
    const float* phi       = (const float*)d_in[8];
    const float* eta       = (const float*)d_in[9];
    const float* CDK       = (const float*)d_in[10];
    const float* CWK       = (const float*)d_in[11];
    const float* CK        = (const float*)d_in[12];

    float* out     = (float*)d_out;
    float* out_eta = out;                       // T*D*K   = 262144
    float* out_phi = out_eta + 262144;          // T*V*K   = 8192000
    float* out_cdk = out_phi + 8192000;         // T*D*K   = 262144
    float* out_cwk = out_cdk + 262144;          // T*V*K   = 8192000
    float* out_ck  = out_cwk + 8192000;         // T*K     = 256
    float* out_z   = out_ck + 256;              // T*D*N   = 1048576
    float* ws      = (float*)d_ws;              // 2*T*K floats = 2 KB

    dtm_init_kernel<<<2048, 256, 0, stream>>>(CWK, CK, out_cwk, out_ck);
    dtm_eta_kernel<<<512, 256, 0, stream>>>(eta, alpha, CDK, noise_eta, out_eta);
    dtm_phistat_kernel<<<16, 256, 0, stream>>>(phi, ws);
    dtm_mh_kernel<<<T_ * D_, N_, 0, stream>>>(W, Z, pw, pt, u_mh, phi, out_eta,
                                              CDK, out_cdk, out_cwk, out_ck, out_z);
    dtm_phi_kernel<<<4000, 256, 0, stream>>>(phi, out_cwk, out_ck, noise_phi, ws, out_phi);
}